// GAT_73349451481180
// MI455X (gfx1250) — compile-verified
//
#include <hip/hip_runtime.h>
#include <math.h>

// GAT layer for MI455X (gfx1250, wave32).
// Sizes fixed to the reference problem.
#define N_NODES 50000
#define N_EDGES 800000
#define D_IN    128
#define N_HEAD  4
#define D_OUT   32            // D_IN / N_HEAD

typedef __attribute__((ext_vector_type(2))) float v2f;
typedef __attribute__((ext_vector_type(8))) float v8f;

// ---- ordered-uint encoding for deterministic float atomicMax ----
__device__ __forceinline__ unsigned f2ord(float f) {
    unsigned u = __float_as_uint(f);
    return (u & 0x80000000u) ? ~u : (u | 0x80000000u);
}
__device__ __forceinline__ float ord2f(unsigned u) {
    return (u & 0x80000000u) ? __uint_as_float(u ^ 0x80000000u)
                             : __uint_as_float(~u);
}

// ---------------------------------------------------------------------------
// K0: init out = bias (broadcast), neigh_sums = 0, gmax = -inf
// ---------------------------------------------------------------------------
__global__ void gat_init_kernel(float* __restrict__ out,
                                const float* __restrict__ bias,
                                float* __restrict__ neigh,
                                unsigned* __restrict__ gmax) {
    int idx = blockIdx.x * blockDim.x + threadIdx.x;
    if (idx < N_NODES * D_IN)
        out[idx] = bias[idx & (D_IN - 1)];
    if (idx < N_NODES * N_HEAD)
        neigh[idx] = 0.0f;
    if (idx == 0)
        *gmax = f2ord(-INFINITY);
}

// ---------------------------------------------------------------------------
// K1: proj = in_feat @ W_proj  via V_WMMA_F32_16X16X4_F32 (fp32 WMMA).
// Block = 256 threads = 8 waves. blockIdx.x -> 16-row tile, wave -> 16-col tile.
// ---------------------------------------------------------------------------
__global__ __launch_bounds__(256)
void gat_gemm_wmma_kernel(const float* __restrict__ A,   // [N_NODES, 128]
                          const float* __restrict__ W,   // [128, 128]
                          float* __restrict__ P) {       // [N_NODES, 128]
    const int wave = threadIdx.x >> 5;
    const int lane = threadIdx.x & 31;
    const int m0   = blockIdx.x * 16;       // row tile (3125 tiles, exact)
    const int n0   = wave * 16;             // col tile (8 waves cover 128)

    const int mRow = m0 + (lane & 15);      // A row for this lane
    const int nCol = n0 + (lane & 15);      // B col for this lane
    const int kGrp = (lane >> 4) * 2;       // K sub-offset: 0 or 2

    v8f acc = {};
#pragma unroll
    for (int k0 = 0; k0 < D_IN; k0 += 4) {
        const int ka = k0 + kGrp;
        v2f a, b;
        a.x = A[mRow * D_IN + ka];
        a.y = A[mRow * D_IN + ka + 1];
        b.x = W[ka * D_IN + nCol];
        b.y = W[(ka + 1) * D_IN + nCol];
        // D = A(16x4) * B(4x16) + C ; 8 args (neg_a, A, neg_b, B, c_mod, C, reuse_a, reuse_b)
        acc = __builtin_amdgcn_wmma_f32_16x16x4_f32(false, a, false, b,
                                                    (short)0, acc, false, false);
    }

    const int mBase = m0 + 8 * (lane >> 4);
#pragma unroll
    for (int v = 0; v < 8; ++v)
        P[(mBase + v) * D_IN + nCol] = acc[v];
}

// ---------------------------------------------------------------------------
// K2: per-node attention scores. Block = 128 threads = 4 waves (one per head).
// Wave h reduces 32 channels with shfl_xor; lane 0 writes score.
// ---------------------------------------------------------------------------
__global__ __launch_bounds__(128)
void gat_scores_kernel(const float* __restrict__ P,
                       const float* __restrict__ a_src,   // [4*32]
                       const float* __restrict__ a_tgt,   // [4*32]
                       float* __restrict__ s_src,         // [N,4]
                       float* __restrict__ s_tgt) {
    const int n = blockIdx.x;
    const int t = threadIdx.x;            // 0..127
    const float p  = P[n * D_IN + t];
    float vs = p * a_src[t];
    float vt = p * a_tgt[t];
#pragma unroll
    for (int off = 16; off > 0; off >>= 1) {
        vs += __shfl_xor(vs, off, 32);
        vt += __shfl_xor(vt, off, 32);
    }
    if ((t & 31) == 0) {
        const int h = t >> 5;
        s_src[n * N_HEAD + h] = vs;
        s_tgt[n * N_HEAD + h] = vt;
    }
}

// ---------------------------------------------------------------------------
// K3: global max of leaky_relu(edge score). One edge per thread, wave-level
// max reduction, one ordered-uint atomicMax per wave.
// ---------------------------------------------------------------------------
__global__ __launch_bounds__(256)
void gat_edge_max_kernel(const int* __restrict__ ei,      // [2, E]
                         const float* __restrict__ s_src,
                         const float* __restrict__ s_tgt,
                         unsigned* __restrict__ gmax) {
    const int e = blockIdx.x * blockDim.x + threadIdx.x;
    float m = -INFINITY;
    if (e < N_EDGES) {
        const int src = ei[e];
        const int tgt = ei[N_EDGES + e];
#pragma unroll
        for (int h = 0; h < N_HEAD; ++h) {
            float s = s_src[src * N_HEAD + h] + s_tgt[tgt * N_HEAD + h];
            s = (s > 0.0f) ? s : 0.2f * s;     // leaky_relu(0.2)
            m = fmaxf(m, s);
        }
    }
#pragma unroll
    for (int off = 16; off > 0; off >>= 1)
        m = fmaxf(m, __shfl_xor(m, off, 32));
    if ((threadIdx.x & 31) == 0)
        atomicMax(gmax, f2ord(m));
}

// ---------------------------------------------------------------------------
// K4: edge_e = exp(score - gmax); neigh_sums[tgt] += edge_e (float atomics,
// 0.8 MB accumulator stays in L2).
// ---------------------------------------------------------------------------
__global__ __launch_bounds__(256)
void gat_edge_exp_kernel(const int* __restrict__ ei,
                         const float* __restrict__ s_src,
                         const float* __restrict__ s_tgt,
                         const unsigned* __restrict__ gmax,
                         float* __restrict__ edge_e,      // [E,4]
                         float* __restrict__ neigh) {     // [N,4]
    const int e = blockIdx.x * blockDim.x + threadIdx.x;
    if (e >= N_EDGES) return;
    const float gm = ord2f(*gmax);
    const int src = ei[e];
    const int tgt = ei[N_EDGES + e];
#pragma unroll
    for (int h = 0; h < N_HEAD; ++h) {
        float s = s_src[src * N_HEAD + h] + s_tgt[tgt * N_HEAD + h];
        s = (s > 0.0f) ? s : 0.2f * s;
        const float ex = __expf(s - gm);
        edge_e[e * N_HEAD + h] = ex;
        atomicAdd(&neigh[tgt * N_HEAD + h], ex);
    }
}

// ---------------------------------------------------------------------------
// K5: aggregation. One thread per (edge, channel): 128 consecutive threads
// share an edge -> fully coalesced 512B gather of proj[src], 512B atomic
// scatter into out[tgt]. proj + out are L2-resident (25.6 MB each, 192 MB L2).
// ---------------------------------------------------------------------------
__global__ __launch_bounds__(256)
void gat_aggregate_kernel(const int* __restrict__ ei,
                          const float* __restrict__ P,
                          const float* __restrict__ edge_e,
                          const float* __restrict__ neigh,
                          float* __restrict__ out) {
    const long long gid = (long long)blockIdx.x * blockDim.x + threadIdx.x;
    if (gid >= (long long)N_EDGES * D_IN) return;
    const int e = (int)(gid >> 7);        // edge
    const int d = (int)(gid & (D_IN - 1));// channel 0..127
    const int h = d >> 5;                 // head
    const int src = ei[e];
    const int tgt = ei[N_EDGES + e];
    const float att = edge_e[e * N_HEAD + h] /
                      (neigh[tgt * N_HEAD + h] + 1e-16f);
    atomicAdd(&out[tgt * D_IN + d], P[src * D_IN + d] * att);
}

// ---------------------------------------------------------------------------
extern "C" void kernel_launch(void* const* d_in, const int* in_sizes, int n_in,
                              void* d_out, int out_size, void* d_ws, size_t ws_size,
                              hipStream_t stream) {
    const float* in_feat = (const float*)d_in[0];
    const int*   ei      = (const int*)  d_in[1];   // [2, E] int32
    const float* W_proj  = (const float*)d_in[3];
    const float* a_src   = (const float*)d_in[4];
    const float* a_tgt   = (const float*)d_in[5];
    const float* bias    = (const float*)d_in[6];
    float*       out     = (float*)d_out;

    // workspace layout
    float*    proj   = (float*)d_ws;                         // N*128
    float*    s_src  = proj  + (size_t)N_NODES * D_IN;       // N*4
    float*    s_tgt  = s_src + (size_t)N_NODES * N_HEAD;     // N*4
    float*    neigh  = s_tgt + (size_t)N_NODES * N_HEAD;     // N*4
    float*    edge_e = neigh + (size_t)N_NODES * N_HEAD;     // E*4
    unsigned* gmax   = (unsigned*)(edge_e + (size_t)N_EDGES * N_HEAD);

    // K0: init outputs / accumulators
    gat_init_kernel<<<(N_NODES * D_IN + 255) / 256, 256, 0, stream>>>(
        out, bias, neigh, gmax);

    // K1: fp32 WMMA projection GEMM (3125 row-tiles x 8 waves)
    gat_gemm_wmma_kernel<<<N_NODES / 16, 256, 0, stream>>>(in_feat, W_proj, proj);

    // K2: per-node scores
    gat_scores_kernel<<<N_NODES, 128, 0, stream>>>(proj, a_src, a_tgt, s_src, s_tgt);

    // K3: global max over edge scores
    gat_edge_max_kernel<<<(N_EDGES + 255) / 256, 256, 0, stream>>>(
        ei, s_src, s_tgt, gmax);

    // K4: exp + segment-sum denominator
    gat_edge_exp_kernel<<<(N_EDGES + 255) / 256, 256, 0, stream>>>(
        ei, s_src, s_tgt, gmax, edge_e, neigh);

    // K5: weighted scatter-add aggregation
    const long long total = (long long)N_EDGES * D_IN;
    gat_aggregate_kernel<<<(unsigned)((total + 255) / 256), 256, 0, stream>>>(
        ei, proj, edge_e, neigh, out);
}